// GaussianSplatter_64888365908303
// MI455X (gfx1250) — compile-verified
//
#include <hip/hip_runtime.h>
#include <math.h>

typedef float v2f __attribute__((ext_vector_type(2)));
typedef float v8f __attribute__((ext_vector_type(8)));

// GCC-style vector type matching the async-to-LDS builtin's parameter type.
typedef int v4i_g __attribute__((vector_size(16)));
typedef __attribute__((address_space(1))) v4i_g gv4i;  // global
typedef __attribute__((address_space(3))) v4i_g lv4i;  // LDS

#define P49   49
#define NPTS  100
#define FEATC 8
#define BB    16
#define HH    112
#define WW    112
#define KPAD  52
#define NPAD  64

// W storage layout: k-pair interleaved so (k, k+1) at column n are 8 contiguous
// bytes -> single aligned ds_load_b64 per B half-fragment.
__host__ __device__ __forceinline__ int widx(int k, int n) {
  return (((k >> 1) * NPAD) + n) * 2 + (k & 1);
}

// ---------------------------------------------------------------------------
// Kernel 1: per-(B,n) plane -> 49 partial bins.  1600 blocks x 128 threads.
// Each thread sums one image row (112 floats, 28 float4 loads) into 7 column
// bins; c = (element index) % 7 is compile-time after unrolling.
// ---------------------------------------------------------------------------
__global__ void k_lbar_partial(const float* __restrict__ logits,
                               float* __restrict__ partial) {
  __shared__ float bins[P49];
  const int bn = blockIdx.x;  // B*100 + n
  if (threadIdx.x < P49) bins[threadIdx.x] = 0.0f;
  __syncthreads();
  const int h = threadIdx.x;
  if (h < HH) {
    const float4* rowv =
        (const float4*)(logits + ((size_t)bn * HH + h) * WW);
    const int r = h % 7;
    float acc[7];
#pragma unroll
    for (int c = 0; c < 7; ++c) acc[c] = 0.0f;
#pragma unroll
    for (int q = 0; q < WW / 4; ++q) {   // 28 float4 loads per row
      const float4 f = rowv[q];
      acc[(4 * q + 0) % 7] += f.x;
      acc[(4 * q + 1) % 7] += f.y;
      acc[(4 * q + 2) % 7] += f.z;
      acc[(4 * q + 3) % 7] += f.w;
    }
#pragma unroll
    for (int c = 0; c < 7; ++c) atomicAdd(&bins[r * 7 + c], acc[c]);
  }
  __syncthreads();
  if (threadIdx.x < P49) partial[bn * P49 + threadIdx.x] = bins[threadIdx.x];
}

// ---------------------------------------------------------------------------
// Kernel 2: reduce partials over the 16 batch images -> lbar[p][n] (49x100).
// ---------------------------------------------------------------------------
__global__ void k_lbar_reduce(const float* __restrict__ partial,
                              float* __restrict__ lbar) {
  const int idx = blockIdx.x * blockDim.x + threadIdx.x;
  if (idx >= P49 * NPTS) return;
  const int n = idx / P49;
  const int p = idx % P49;
  float s = 0.0f;
  for (int b = 0; b < BB; ++b) s += partial[(b * NPTS + n) * P49 + p];
  lbar[p * NPTS + n] = s * (1.0f / 4096.0f);
}

// ---------------------------------------------------------------------------
// Kernel 3: build the 49x49 weight matrix W into the interleaved padded
// 52x64 layout.  One block, 64 threads; threads 0..48 own one position p.
// ---------------------------------------------------------------------------
__device__ __forceinline__ float samp7(const float k7[7][7], float iz, float jz) {
  const bool valid = (iz >= 0.0f) && (iz <= 6.0f) && (jz >= 0.0f) && (jz <= 6.0f);
  const int ic = (int)fminf(fmaxf(iz, 0.0f), 6.0f);
  const int jc = (int)fminf(fmaxf(jz, 0.0f), 6.0f);
  const float v = k7[ic][jc];
  return valid ? v : 0.0f;
}

__global__ void k_build_w(const float* __restrict__ lbar,
                          const float* __restrict__ sigma_x,
                          const float* __restrict__ sigma_y,
                          const float* __restrict__ opacity,
                          const float* __restrict__ rho,
                          float* __restrict__ wpad) {
  for (int i = threadIdx.x; i < KPAD * NPAD; i += blockDim.x) wpad[i] = 0.0f;
  __syncthreads();
  const int p = threadIdx.x;
  if (p >= P49) return;

  float wsx = 0.0f, wsy = 0.0f, wop = 0.0f, wrho = 0.0f;
  for (int n = 0; n < NPTS; ++n) {
    const float l = lbar[p * NPTS + n];
    wsx  += l * sigma_x[n];
    wsy  += l * sigma_y[n];
    wop  += l * opacity[n];
    wrho += l * rho[n];
  }
  const float a   = wsx * wsx + 1e-5f;
  const float d   = wsy * wsy + 1e-5f;
  const float b   = wrho * wsx * wsy;
  const float det = a * d - b * b;
  const float ia = d / det, ib = -b / det, idd = a / det;
  const float norm = 1.0f / (2.0f * 3.14159265358979323846f * sqrtf(det));

  float k7[7][7];
#pragma unroll
  for (int i = 0; i < 7; ++i)
#pragma unroll
    for (int j = 0; j < 7; ++j) k7[i][j] = 0.0f;

  float k5[5][5];
  float kmax = 0.0f;
#pragma unroll
  for (int i = 0; i < 5; ++i) {
    const float x = -5.0f + 2.5f * (float)i;
#pragma unroll
    for (int j = 0; j < 5; ++j) {
      const float y = -5.0f + 2.5f * (float)j;
      const float z = -0.5f * (ia * x * x + 2.0f * ib * x * y + idd * y * y);
      const float v = expf(z) * norm;
      k5[i][j] = v;
      kmax = fmaxf(kmax, v);
    }
  }
  const float inv = 1.0f / kmax;
#pragma unroll
  for (int i = 0; i < 5; ++i)
#pragma unroll
    for (int j = 0; j < 5; ++j) k7[i + 1][j + 1] = k5[i][j] * inv;

  const int r = p / 7, c = p % 7;
  const float sx = (1.0f - 2.0f * (float)c / 7.0f) * 3.0f;  // tx * (COL-1)/2
  const float sy = (1.0f - 2.0f * (float)r / 7.0f) * 3.0f;  // ty * (ROW-1)/2

  for (int i = 0; i < 7; ++i) {
    const float ii = (float)i + sy;
    const float i0 = floorf(ii);
    const float wi = ii - i0;
    for (int j = 0; j < 7; ++j) {
      const float jj = (float)j + sx;
      const float j0 = floorf(jj);
      const float wj = jj - j0;
      const float v00 = samp7(k7, i0,        j0);
      const float v01 = samp7(k7, i0,        j0 + 1.0f);
      const float v10 = samp7(k7, i0 + 1.0f, j0);
      const float v11 = samp7(k7, i0 + 1.0f, j0 + 1.0f);
      const float val = v00 * (1.0f - wi) * (1.0f - wj) + v01 * (1.0f - wi) * wj +
                        v10 * wi * (1.0f - wj)          + v11 * wi * wj;
      wpad[widx(p, i * 7 + j)] = wop * val;
    }
  }
}

// ---------------------------------------------------------------------------
// Kernel 4: WMMA GEMM.  out[(patch,ch), ij] = sum_p A[(patch,ch),p] * W[p,ij]
// M = 32768 (2048 tiles of 16), K = 49 -> 52, N = 49 -> 64 (4 tiles of 16).
// 256 blocks x 256 threads (8 waves); one m-tile per wave.
// W staged into LDS via gfx1250 async-to-LDS copies when available.
// ---------------------------------------------------------------------------
__global__ void __launch_bounds__(256)
k_splat_wmma(const float* __restrict__ inp, const float* __restrict__ wpad,
             float* __restrict__ out) {
  __shared__ float wlds[KPAD * NPAD];

#if __has_builtin(__builtin_amdgcn_global_load_async_to_lds_b128)
  for (int c4 = threadIdx.x; c4 < (KPAD * NPAD) / 4; c4 += blockDim.x) {
    __builtin_amdgcn_global_load_async_to_lds_b128(
        (gv4i*)(wpad) + c4,
        (lv4i*)(&wlds[0]) + c4,
        /*offset=*/0, /*cpol=*/0);
  }
#if __has_builtin(__builtin_amdgcn_s_wait_asynccnt)
  __builtin_amdgcn_s_wait_asynccnt(0);
#else
  asm volatile("s_wait_asynccnt 0x0" ::: "memory");
#endif
#else
  for (int i = threadIdx.x; i < KPAD * NPAD; i += blockDim.x) wlds[i] = wpad[i];
#endif
  __syncthreads();

  const int lane  = threadIdx.x & 31;
  const int wave  = threadIdx.x >> 5;
  const int mtile = blockIdx.x * 8 + wave;       // 0..2047
  const int col   = lane & 15;
  const int koff  = (lane >> 4) << 1;            // 0 or 2 (A/B K sub-offset)

  // A fragment source row for this lane
  const int m     = mtile * 16 + col;            // global GEMM row
  const int patch = m >> 3;                      // 0..4095
  const int ch    = m & 7;
  const int bi    = patch >> 8;                  // /256
  const int rem   = patch & 255;
  const int nh    = rem >> 4;
  const int nw    = rem & 15;
  const float* abase =
      inp + (((size_t)(bi * 64 + ch) * HH) + (size_t)nh * 7) * WW + nw * 7;

  const v2f* __restrict__ wv = (const v2f*)wlds;   // interleaved pairs
  const int wrow = koff >> 1;                      // 0 or 1

  v8f acc0 = {}, acc1 = {}, acc2 = {}, acc3 = {};
#pragma unroll
  for (int k = 0; k < KPAD; k += 4) {
    const int p0 = k + koff;
    const int p1 = p0 + 1;
    v2f a;
    a.x = (p0 < P49) ? abase[(p0 / 7) * WW + (p0 % 7)] : 0.0f;
    a.y = (p1 < P49) ? abase[(p1 / 7) * WW + (p1 % 7)] : 0.0f;
    const int wb = ((k >> 1) + wrow) * NPAD + col;
    const v2f b0 = wv[wb];
    const v2f b1 = wv[wb + 16];
    const v2f b2 = wv[wb + 32];
    const v2f b3 = wv[wb + 48];
    acc0 = __builtin_amdgcn_wmma_f32_16x16x4_f32(false, a, false, b0, (short)0, acc0, false, false);
    acc1 = __builtin_amdgcn_wmma_f32_16x16x4_f32(false, a, false, b1, (short)0, acc1, false, false);
    acc2 = __builtin_amdgcn_wmma_f32_16x16x4_f32(false, a, false, b2, (short)0, acc2, false, false);
    acc3 = __builtin_amdgcn_wmma_f32_16x16x4_f32(false, a, false, b3, (short)0, acc3, false, false);
  }

  // Scatter D: VGPR r holds rows {r, r+8}; lanes 0-15 -> N=lane, 16-31 -> N=lane-16
  const int hi8 = (lane >> 4) << 3;
  int oi[4], oj[4];
  bool nok[4];
#pragma unroll
  for (int t = 0; t < 4; ++t) {
    const int n = t * 16 + col;
    nok[t] = (n < P49);
    oi[t] = n / 7;
    oj[t] = n % 7;
  }
#pragma unroll
  for (int r = 0; r < 8; ++r) {
    const int mr     = mtile * 16 + r + hi8;
    const int patch2 = mr >> 3;
    const int ch2    = mr & 7;
    const int bi2    = patch2 >> 8;
    const int rem2   = patch2 & 255;
    const int nh2    = rem2 >> 4;
    const int nw2    = rem2 & 15;
    float* obase =
        out + (((size_t)(bi2 * FEATC + ch2) * HH) + (size_t)nh2 * 7) * WW + nw2 * 7;
    if (nok[0]) obase[oi[0] * WW + oj[0]] = acc0[r];
    if (nok[1]) obase[oi[1] * WW + oj[1]] = acc1[r];
    if (nok[2]) obase[oi[2] * WW + oj[2]] = acc2[r];
    if (nok[3]) obase[oi[3] * WW + oj[3]] = acc3[r];
  }
}

// ---------------------------------------------------------------------------
extern "C" void kernel_launch(void* const* d_in, const int* in_sizes, int n_in,
                              void* d_out, int out_size, void* d_ws, size_t ws_size,
                              hipStream_t stream) {
  (void)in_sizes; (void)n_in; (void)out_size; (void)ws_size;
  const float* inp    = (const float*)d_in[0];
  const float* logits = (const float*)d_in[1];
  const float* sigx   = (const float*)d_in[2];
  const float* sigy   = (const float*)d_in[3];
  const float* opac   = (const float*)d_in[4];
  const float* rho    = (const float*)d_in[5];
  float* out = (float*)d_out;

  float* ws      = (float*)d_ws;
  float* partial = ws;            // 1600*49 = 78400 floats
  float* lbar    = ws + 80000;    // 49*100  =  4900 floats
  float* wpad    = ws + 88000;    // 52*64   =  3328 floats

  k_lbar_partial<<<BB * NPTS, 128, 0, stream>>>(logits, partial);
  k_lbar_reduce<<<(P49 * NPTS + 255) / 256, 256, 0, stream>>>(partial, lbar);
  k_build_w<<<1, 64, 0, stream>>>(lbar, sigx, sigy, opac, rho, wpad);
  k_splat_wmma<<<256, 256, 0, stream>>>(inp, wpad, out);
}